// PSMCosineModule_31980326486034
// MI455X (gfx1250) — compile-verified
//
#include <hip/hip_runtime.h>

// PSM cosine cost volume on MI455X (gfx1250), fp32-exact via V_WMMA_F32_16X16X4_F32.
// out[b,d,h,w] = (1/512) * sum_c L[b,c,h,w] * R[b,c,h,w-d], zero for w < d.
//
// Banded-Gram tiling: one wave owns a 16-wide left tile and 4 accumulator
// blocks (right tiles at offsets 0,-16,-32,-48). Invalid right columns are
// clamp-loaded unmasked; since okB depends only on (lane&15) and D-matrix
// column N lives in lanes with lane&15==N, the garbage is confined to lanes
// we zero at store time (which is exactly the reference's zero padding).
//
// K loop steps 16 channels per iteration with compile-time-constant channel
// offsets so all load offsets fold into the 24-bit instruction immediate
// (max 13*HW*4 = 1.56 MB < 8.3 MB); only 5 pointer adds per macro-step.

typedef float v2f __attribute__((ext_vector_type(2)));
typedef float v8f __attribute__((ext_vector_type(8)));

#define BATCH 4
#define CH    512
#define HGT   96
#define WID   312
#define DISP  48
#define HW    (HGT * WID)   // 29952 elements per channel plane
#define NTILE 20            // ceil(312 / 16)

__global__ __launch_bounds__(128) void psm_cost_wmma(
    const float* __restrict__ L,
    const float* __restrict__ R,
    float* __restrict__ out)
{
    const int lane = threadIdx.x & 31;
    const int wv   = threadIdx.x >> 5;          // 4 waves per block
    const int t    = blockIdx.x * 4 + wv;       // left tile index 0..19
    const int h    = blockIdx.y;
    const int b    = blockIdx.z;
    const int w0   = t * 16;

    const int half = lane >> 4;                 // 0: K={0,1}; 1: K={2,3} (A/B frag layout)
    const int lk   = lane & 15;                 // row (A) / col (B) within 16

    // Uniform (SGPR) row bases; per-lane offsets stay 32-bit so loads use the
    // saddr + voffset (scale_offset) global-load form.
    const float* __restrict__ Lrow = L + (size_t)b * CH * HW + (size_t)h * WID;
    const float* __restrict__ Rrow = R + (size_t)b * CH * HW + (size_t)h * WID;

    // ---- A fragment offset: left column w0+lk (clamped rows are never stored) ----
    int wA = w0 + lk;
    if (wA >= WID) wA = WID - 1;
    unsigned offA = (unsigned)(2 * half * HW + wA);

    // ---- B fragment offsets: right columns w0 - 16j + lk (clamped, unmasked) ----
    unsigned offB[4];
    bool okB[4];
#pragma unroll
    for (int j = 0; j < 4; ++j) {
        int wB = w0 - 16 * j + lk;
        okB[j] = (wB >= 0);                     // w' < 0  <=>  d > w  => output must be 0
        if (wB < 0) wB = 0;                     // clamp: garbage confined to lane lk
        if (wB >= WID) wB = WID - 1;            // (w' >= WID feeds only non-stored elems)
        offB[j] = (unsigned)(2 * half * HW + wB);
    }

    v8f acc[4];
#pragma unroll
    for (int j = 0; j < 4; ++j) acc[j] = 0.0f;

    // ---- K loop: 16 channels (4 WMMA K-steps) per iteration ----
    for (int kk = 0; kk < CH; kk += 16) {
#pragma unroll
        for (int c = 0; c < 4; ++c) {
            const unsigned co = (unsigned)(4 * c) * HW;   // compile-time constant

            v2f a;
            a.x = Lrow[offA + co];
            a.y = Lrow[offA + co + HW];

            v2f bf[4];
#pragma unroll
            for (int j = 0; j < 4; ++j) {
                bf[j].x = Rrow[offB[j] + co];
                bf[j].y = Rrow[offB[j] + co + HW];
            }

#pragma unroll
            for (int j = 0; j < 4; ++j) {
                acc[j] = __builtin_amdgcn_wmma_f32_16x16x4_f32(
                    /*neg_a=*/false, a, /*neg_b=*/false, bf[j],
                    /*c_mod=*/(short)0, acc[j], /*reuse_a=*/false, /*reuse_b=*/false);
            }
        }

        offA += 16u * HW;
#pragma unroll
        for (int j = 0; j < 4; ++j) offB[j] += 16u * HW;
    }

    // ---- Scatter the banded results: d = 16j + M - lk, w = w0 + M ----
    const float scale = 1.0f / (float)CH;
#pragma unroll
    for (int j = 0; j < 4; ++j) {
#pragma unroll
        for (int v = 0; v < 8; ++v) {
            const int M = v + 8 * half;         // D row: lane<16 -> v, lane>=16 -> v+8
            const int w = w0 + M;
            const int d = 16 * j + M - lk;
            if (d >= 0 && d < DISP && w < WID) {
                const float val = okB[j] ? acc[j][v] * scale : 0.0f;
                out[(((size_t)b * DISP + d) * HGT + h) * WID + w] = val;
            }
        }
    }
}

extern "C" void kernel_launch(void* const* d_in, const int* in_sizes, int n_in,
                              void* d_out, int out_size, void* d_ws, size_t ws_size,
                              hipStream_t stream) {
    const float* L = (const float*)d_in[0];
    const float* R = (const float*)d_in[1];
    float* out = (float*)d_out;

    dim3 grid(NTILE / 4, HGT, BATCH);   // (5, 96, 4) blocks, 4 waves each -> 7680 waves
    dim3 block(128);
    psm_cost_wmma<<<grid, block, 0, stream>>>(L, R, out);
}